// ShortestPathLoss_82927228551954
// MI455X (gfx1250) — compile-verified
//
#include <hip/hip_runtime.h>

#define C_DIM 4096
#define BLOCK 512
#define CHUNK 256            // elements per wave (BLOCK/32 waves * CHUNK = C_DIM)

typedef unsigned int u32x4 __attribute__((ext_vector_type(4)));
typedef int          i32x4 __attribute__((ext_vector_type(4)));
typedef int          i32x8 __attribute__((ext_vector_type(8)));

// Monotonic float->uint map + index packing for stable *descending* argsort:
// larger float => larger key; equal floats => smaller index has larger key.
__device__ __forceinline__ unsigned long long pack_key(float v, unsigned int idx) {
  unsigned int u = __float_as_uint(v);
  u = (u & 0x80000000u) ? ~u : (u | 0x80000000u);
  return ((unsigned long long)u << 32) | (unsigned long long)(~idx);
}

// One bitonic substage (width j, stage k) restricted to this wave's 256-element
// chunk. Same-wave LDS ops execute in order at the LDS, so no s_barrier is
// needed; the wave_barrier() only stops the compiler from reordering.
__device__ __forceinline__ void wave_local_pass(unsigned long long* key,
                                                int base, int j, int k, int lane) {
  for (int p = lane; p < CHUNK / 2; p += 32) {
    int i = base + (((p & ~(j - 1)) << 1) | (p & (j - 1)));
    int l = i | j;
    unsigned long long a = key[i];
    unsigned long long c = key[l];
    bool desc = ((i & k) == 0);
    if (desc ? (a < c) : (a > c)) { key[i] = c; key[l] = a; }
  }
  __builtin_amdgcn_wave_barrier();
}

__global__ __launch_bounds__(BLOCK) void spl_row_kernel(
    const float* __restrict__ logits,
    const float* __restrict__ P,
    const int*   __restrict__ labels,
    float*       __restrict__ partial)
{
  extern __shared__ unsigned char smem[];
  unsigned long long* key  = (unsigned long long*)smem;                 // 32 KB
  float*              prow = (float*)(smem + (size_t)C_DIM * 8);        // 16 KB
  __shared__ float red[BLOCK / 32];

  const int b    = blockIdx.x;
  const int tid  = threadIdx.x;
  const int wave = tid >> 5;
  const int lane = tid & 31;
  const int base = wave * CHUNK;
  const int t    = labels[b];

#if __has_builtin(__builtin_amdgcn_tensor_load_to_lds)
  // --- TDM: DMA P[t, 0:4096] (16 KB contiguous) into LDS, async vs. the sort.
  // Default (RT) cache policy: P rows are re-read by other blocks -> keep in L2.
  if (tid == 0) {
    unsigned long long gaddr = (unsigned long long)(P + (size_t)t * C_DIM);
    unsigned int lds_addr    = (unsigned int)(size_t)prow;
    u32x4 g0;
    g0.x = 1u;                                              // count=1, user mode
    g0.y = lds_addr;                                        // LDS dest (bytes)
    g0.z = (unsigned int)(gaddr & 0xFFFFFFFFull);           // global_addr[31:0]
    g0.w = (unsigned int)((gaddr >> 32) & 0x01FFFFFFull)    // global_addr[56:32]
         | 0x80000000u;                                     // type=2 ("image")
    i32x8 g1;
    g1[0] = 0x00020000;                                     // data_size=2 (4 B)
    g1[1] = (int)((C_DIM & 0xFFFF) << 16);                  // tensor_dim0 lo16
    g1[2] = (int)(((C_DIM >> 16) & 0xFFFF) | (1 << 16));    // dim0 hi16 | tensor_dim1=1
    g1[3] = (int)(C_DIM << 16);                             // tile_dim0 = 4096
    g1[4] = 1;                                              // tile_dim1 = 1
    g1[5] = C_DIM;                                          // tensor_dim0_stride lo32
    g1[6] = 0;
    g1[7] = 0;
    i32x4 gz = {0, 0, 0, 0};                                // groups 2/3 unused (<=2D)
#if defined(__clang_major__) && (__clang_major__ >= 23)
    i32x8 gz8 = {0, 0, 0, 0, 0, 0, 0, 0};
    __builtin_amdgcn_tensor_load_to_lds(g0, g1, gz, gz, gz8, 0);
#else
    __builtin_amdgcn_tensor_load_to_lds(g0, g1, gz, gz, 0);
#endif
  }
#else
  // Fallback: cooperative vector load of the P row into LDS.
  for (int i = tid; i < C_DIM; i += BLOCK)
    prow[i] = P[(size_t)t * C_DIM + i];
#endif

  // --- build keys for this wave's own chunk (coalesced, streamed NT so the
  // 128 MB logits stream does not evict P from L2) ---
  const float* rowp = logits + (size_t)b * C_DIM;
  for (int i = base + lane; i < base + CHUNK; i += 32)
    key[i] = pack_key(__builtin_nontemporal_load(rowp + i), (unsigned int)i);
  __builtin_amdgcn_wave_barrier();

  // --- bitonic sort, descending ---
  // Phase 1: stages k=2..CHUNK are entirely wave-local (j<=128): no s_barrier.
  for (int k = 2; k <= CHUNK; k <<= 1)
    for (int j = k >> 1; j > 0; j >>= 1)
      wave_local_pass(key, base, j, k, lane);

  // Phase 2: stages k=512..4096. Only j>=CHUNK substages cross chunks and
  // need full-block barriers (10 total); the j<=128 tail is wave-local again.
  for (int k = 2 * CHUNK; k <= C_DIM; k <<= 1) {
    for (int j = k >> 1; j >= CHUNK; j >>= 1) {
      __syncthreads();
      for (int p = tid; p < C_DIM / 2; p += BLOCK) {
        int i = ((p & ~(j - 1)) << 1) | (p & (j - 1));
        int l = i | j;
        unsigned long long a = key[i];
        unsigned long long c = key[l];
        bool desc = ((i & k) == 0);
        if (desc ? (a < c) : (a > c)) { key[i] = c; key[l] = a; }
      }
    }
    __syncthreads();
    for (int j = CHUNK / 2; j > 0; j >>= 1)
      wave_local_pass(key, base, j, k, lane);
  }
  __syncthreads();

#if __has_builtin(__builtin_amdgcn_tensor_load_to_lds)
  // TDM had the entire sort to complete; drain TENSORcnt (no-op for waves
  // that issued nothing), then publish the LDS tile to all waves.
  __builtin_amdgcn_s_wait_tensorcnt(0);
  __syncthreads();
#endif

  // --- acc = sum_j P[t, order[j]] / (j+1), gathered from LDS ---
  float acc = 0.0f;
  for (int jj = tid; jj < C_DIM; jj += BLOCK) {
    unsigned int idx = ~(unsigned int)(key[jj] & 0xFFFFFFFFull);
    acc += prow[idx & (C_DIM - 1)] * (1.0f / (float)(jj + 1));
  }

  // --- block reduction (wave32 shuffles + LDS) ---
  for (int o = 16; o > 0; o >>= 1) acc += __shfl_down(acc, o, 32);
  if (lane == 0) red[wave] = acc;
  __syncthreads();
  if (tid < 32) {
    float v = (tid < BLOCK / 32) ? red[tid] : 0.0f;
    for (int o = 16; o > 0; o >>= 1) v += __shfl_down(v, o, 32);
    if (tid == 0) partial[b] = v;
  }
}

__global__ __launch_bounds__(1024) void spl_reduce_kernel(
    const float* __restrict__ partial, float* __restrict__ out,
    int n, float scale)
{
  __shared__ float red[32];
  float acc = 0.0f;
  for (int i = threadIdx.x; i < n; i += 1024) acc += partial[i];
  for (int o = 16; o > 0; o >>= 1) acc += __shfl_down(acc, o, 32);
  if ((threadIdx.x & 31) == 0) red[threadIdx.x >> 5] = acc;
  __syncthreads();
  if (threadIdx.x < 32) {
    float v = red[threadIdx.x];
    for (int o = 16; o > 0; o >>= 1) v += __shfl_down(v, o, 32);
    if (threadIdx.x == 0) out[0] = v * scale;
  }
}

extern "C" void kernel_launch(void* const* d_in, const int* in_sizes, int n_in,
                              void* d_out, int out_size, void* d_ws, size_t ws_size,
                              hipStream_t stream) {
  const float* logits = (const float*)d_in[0];   // (B, 4096) f32
  const float* P      = (const float*)d_in[1];   // (4096, 4096) f32
  const int*   labels = (const int*)d_in[2];     // (B,) int
  const int B = in_sizes[2];                     // 8192

  float* partial = (float*)d_ws;                 // B floats of scratch

  const size_t smem = (size_t)C_DIM * 8 + (size_t)C_DIM * 4;  // keys + P row = 48 KB
  spl_row_kernel<<<B, BLOCK, smem, stream>>>(logits, P, labels, partial);
  spl_reduce_kernel<<<1, 1024, 0, stream>>>(partial, (float*)d_out, B, 1.0f / (float)B);
}